// MPNN_40080634806394
// MI455X (gfx1250) — compile-verified
//
#include <hip/hip_runtime.h>

// Problem constants (fixed by the reference)
#define BATCH   32
#define U1SZ    61
#define IVAL    40
#define NNODES  (BATCH * (U1SZ + IVAL))   // 3232
#define NEDGE   32768
#define D64     64
#define TCOLS   192                       // [ WL | BL | root ] fused

typedef float v2f __attribute__((ext_vector_type(2)));
typedef float v8f __attribute__((ext_vector_type(8)));

// ---------------------------------------------------------------- embed
// H[n][d] = x[n] * W_u[d] + b_u[d]   (concat(x_u,x_v) == x flattened)
__global__ __launch_bounds__(256) void mpnn_embed(
    const float* __restrict__ x, const float* __restrict__ Wu,
    const float* __restrict__ bu, float* __restrict__ H) {
  int t = blockIdx.x * 256 + threadIdx.x;           // N*64 threads exactly
  int n = t >> 6, d = t & 63;
  H[t] = fmaf(x[n], Wu[d], bu[d]);
}

// ---------------------------------------------------------------- weight pack
// Wcat[k][0:64]=WL[k][:], [64:128]=BLmat[k][:], [128:192]=root[k][:]
__global__ __launch_bounds__(256) void mpnn_pack_w(
    const float* __restrict__ Wl, const float* __restrict__ bl,
    const float* __restrict__ root, float* __restrict__ Wcat) {
  int t = blockIdx.x * 256 + threadIdx.x;
  if (t >= D64 * TCOLS) return;
  int k = t / TCOLS, j = t % TCOLS;
  float v;
  if (j < 64)        v = Wl[k * 64 + j];
  else if (j < 128)  v = bl[k * 64 + (j - 64)];
  else               v = root[k * 64 + (j - 128)];
  Wcat[t] = v;
}

// ---------------------------------------------------------------- WMMA GEMM
// T[3232 x 192] = H[3232 x 64] @ Wcat[64 x 192], fp32 via v_wmma_f32_16x16x4_f32.
// One wave per 16x16 output tile; 16 chained WMMAs cover K=64.
__global__ __launch_bounds__(128) void mpnn_gemm_wmma(
    const float* __restrict__ H, const float* __restrict__ W,
    float* __restrict__ T) {
  const int wave = threadIdx.x >> 5;
  const int lane = threadIdx.x & 31;
  const int tile = blockIdx.x * 4 + wave;           // 2424 tiles exactly
  const int tm = tile / (TCOLS / 16);               // 0..201
  const int tn = tile % (TCOLS / 16);               // 0..11
  const int row0 = tm * 16, col0 = tn * 16;

  const int mn = lane & 15;                         // M index for A, N index for B
  const int kk = (lane >> 4) << 1;                  // lanes 0-15 -> K{0,1}; 16-31 -> K{2,3}
  const float* hp = H + (row0 + mn) * D64 + kk;     // A: 16x4 f32 fragment base
  const float* wp = W + kk * TCOLS + col0 + mn;     // B: 4x16 f32 fragment base

  v8f acc = {};
#pragma unroll
  for (int k0 = 0; k0 < D64; k0 += 4) {
    v2f a, b;
    a.x = hp[k0];
    a.y = hp[k0 + 1];
    b.x = wp[k0 * TCOLS];
    b.y = wp[(k0 + 1) * TCOLS];
    // (neg_a, A, neg_b, B, c_mod, C, reuse_a, reuse_b)
    acc = __builtin_amdgcn_wmma_f32_16x16x4_f32(false, a, false, b,
                                                (short)0, acc, false, false);
  }

  // D layout: VGPR v holds M=v (lanes 0-15) / M=v+8 (lanes 16-31), N=lane&15
  const int mbase = row0 + ((lane >> 4) << 3);
#pragma unroll
  for (int v = 0; v < 8; ++v)
    T[(mbase + v) * TCOLS + col0 + mn] = acc[v];
}

// ---------------------------------------------------------------- zero scratch
__global__ __launch_bounds__(256) void mpnn_zero(float* __restrict__ p, int n) {
  int t = blockIdx.x * 256 + threadIdx.x;
  if (t < n) p[t] = 0.0f;
}

// ---------------------------------------------------------------- edge scatter
// msg_e[d] = ea_e * A[src][d] + Bm[src][d]; atomic scatter-add into dst + count.
__global__ __launch_bounds__(256) void mpnn_edges(
    const int* __restrict__ ei, const float* __restrict__ ea,
    const float* __restrict__ T, float* __restrict__ summed,
    float* __restrict__ cnt) {
  int t = blockIdx.x * 256 + threadIdx.x;           // NEDGE*64 threads exactly
  int e = t >> 6, d = t & 63;
  int s   = ei[e];
  int dst = ei[NEDGE + e];
  float a = ea[e];
  const float* ts = T + s * TCOLS;
  float m = fmaf(a, ts[d], ts[64 + d]);             // ea*(H@WL) + (H@BL)
  atomicAdd(summed + dst * D64 + d, m);             // global_atomic_add_f32
  if (d == 0) atomicAdd(cnt + dst, 1.0f);
}

// ---------------------------------------------------------------- finalize
// out = summed/max(cnt,1) + (H@root) + bias  (+ReLU on layer 1)
__global__ __launch_bounds__(256) void mpnn_finalize(
    const float* __restrict__ summed, const float* __restrict__ cnt,
    const float* __restrict__ T, const float* __restrict__ bias,
    float* __restrict__ out, int relu) {
  int t = blockIdx.x * 256 + threadIdx.x;           // N*64 threads exactly
  int n = t >> 6, d = t & 63;
  float v = summed[t] / fmaxf(cnt[n], 1.0f) + T[n * TCOLS + 128 + d] + bias[d];
  if (relu) v = fmaxf(v, 0.0f);
  out[t] = v;
}

// ---------------------------------------------------------------- launch
extern "C" void kernel_launch(void* const* d_in, const int* in_sizes, int n_in,
                              void* d_out, int out_size, void* d_ws, size_t ws_size,
                              hipStream_t stream) {
  const float* x     = (const float*)d_in[0];
  const float* Wu    = (const float*)d_in[1];
  const float* bu    = (const float*)d_in[2];
  const float* Wl1   = (const float*)d_in[3];
  const float* bl1   = (const float*)d_in[4];
  const float* Wl2   = (const float*)d_in[5];
  const float* bl2   = (const float*)d_in[6];
  const float* root1 = (const float*)d_in[7];
  const float* bias1 = (const float*)d_in[8];
  const float* root2 = (const float*)d_in[9];
  const float* bias2 = (const float*)d_in[10];
  const int*   ei    = (const int*)d_in[11];
  const float* eattr = (const float*)d_in[12];
  float*       out   = (float*)d_out;

  // Workspace layout (floats): ~5.0 MB total
  float* ws  = (float*)d_ws;
  float* H   = ws;                       // N*64
  float* T   = H  + NNODES * D64;        // N*192
  float* SUM = T  + NNODES * TCOLS;      // N*64   (SUM and CNT contiguous)
  float* CNT = SUM + NNODES * D64;       // N
  float* WC  = CNT + NNODES;             // 64*192
  float* H2  = WC + D64 * TCOLS;         // N*64

  const int nd       = NNODES * D64;               // 206848
  const int nd_blk   = nd / 256;                   // 808 exact
  const int zero_n   = nd + NNODES;                // SUM + CNT
  const int zero_blk = (zero_n + 255) / 256;
  const int pack_blk = (D64 * TCOLS + 255) / 256;
  const int gemm_blk = (NNODES / 16) * (TCOLS / 16) / 4;  // 606 exact
  const int edge_blk = (NEDGE * D64) / 256;               // 8192 exact

  mpnn_embed<<<nd_blk, 256, 0, stream>>>(x, Wu, bu, H);

  // ---- layer 1 (ReLU) ----
  mpnn_pack_w<<<pack_blk, 256, 0, stream>>>(Wl1, bl1, root1, WC);
  mpnn_gemm_wmma<<<gemm_blk, 128, 0, stream>>>(H, WC, T);
  mpnn_zero<<<zero_blk, 256, 0, stream>>>(SUM, zero_n);
  mpnn_edges<<<edge_blk, 256, 0, stream>>>(ei, eattr, T, SUM, CNT);
  mpnn_finalize<<<nd_blk, 256, 0, stream>>>(SUM, CNT, T, bias1, H2, 1);

  // ---- layer 2 (no ReLU) ----
  mpnn_pack_w<<<pack_blk, 256, 0, stream>>>(Wl2, bl2, root2, WC);
  mpnn_gemm_wmma<<<gemm_blk, 128, 0, stream>>>(H2, WC, T);
  mpnn_zero<<<zero_blk, 256, 0, stream>>>(SUM, zero_n);
  mpnn_edges<<<edge_blk, 256, 0, stream>>>(ei, eattr, T, SUM, CNT);
  mpnn_finalize<<<nd_blk, 256, 0, stream>>>(SUM, CNT, T, bias2, out, 0);
}